// Lingunet5GroupFilter_49469433315713
// MI455X (gfx1250) — compile-verified
//
#include <hip/hip_runtime.h>
#include <hip/hip_bf16.h>

typedef __attribute__((ext_vector_type(2))) float v2f;
typedef __attribute__((ext_vector_type(8))) float v8f;

#define EPS 1e-5f

// ---------------------------------------------------------------------------
// Zero the stats workspace (4 layers * 2048 (b,o) pairs * {sum, sumsq}).
// ---------------------------------------------------------------------------
__global__ void zero_stats(float* __restrict__ stats, int n) {
    int i = blockIdx.x * blockDim.x + threadIdx.x;
    if (i < n) stats[i] = 0.0f;
}

// ---------------------------------------------------------------------------
// Batched GEMM  Y[b] = W[b] (16x128 tile) x X[b] (128 x HW) using
// V_WMMA_F32_16X16X4_F32, plus fused accumulation of per-row sum / sumsq
// for the instance norm (atomics into `stats`).
//
// Block: 128 threads = 4 waves. Each wave owns a 16(M) x 64(N) stripe.
// Grid: (HW/256 (ceil), O/16, B).  HW is a template constant so all row
// strides fold into the 24-bit instruction offsets.
//
// FULL=true  -> grid tiles HW exactly (HW >= 256 here): no bounds handling.
// FULL=false -> out-of-range tiles get clamped load addresses, masked
//               stats contributions, and skipped stores (HW = 64).
// ---------------------------------------------------------------------------
template<int HW, bool DOSTATS, bool FULL>
__global__ __launch_bounds__(128)
void dynconv_gemm(const float* __restrict__ x,      // [B,128,HW]
                  const float* __restrict__ wt,     // [B,128,128]
                  float* __restrict__ y,            // [B,128,HW]
                  float* __restrict__ stats)        // [2048][2] (sum, sumsq)
{
    __shared__ float ldsW[16][132];                 // padded: stride%64==4 -> no bank conflicts

    const int b   = blockIdx.z;
    const int o0  = blockIdx.y * 16;
    const int pB  = blockIdx.x * 256;
    const int tid = threadIdx.x;

    // ---- cooperative load of the 16x128 fp32 weight tile into LDS ----
    const float* wsrc = wt + (size_t)(b * 128 + o0) * 128;   // rows contiguous
    #pragma unroll
    for (int i = 0; i < 16; ++i) {
        int idx = tid + i * 128;                   // 0..2047
        ldsW[idx >> 7][idx & 127] = wsrc[idx];
    }
    __syncthreads();

    const int lane = tid & 31;
    const int wv   = tid >> 5;
    const int m16  = lane & 15;                    // row (A) / column (B) within tile
    const int hi   = lane >> 4;                    // 0: K=0..1 / rows 0..7, 1: K=2..3 / rows 8..15
    const int koff = hi * 2;

    const float* xb = x + (size_t)b * 128 * HW;
    const int pw = pB + wv * 64;

    const int col0 = pw +  0 + m16;
    const int col1 = pw + 16 + m16;
    const int col2 = pw + 32 + m16;
    const int col3 = pw + 48 + m16;

    // tile validity (uniform within the wave; compile-time true when FULL)
    const bool tv0 = FULL || (pw +  0) < HW;
    const bool tv1 = FULL || (pw + 16) < HW;
    const bool tv2 = FULL || (pw + 32) < HW;
    const bool tv3 = FULL || (pw + 48) < HW;

    // load pointers (clamped into range when !FULL; HW >= 64 > 15)
    const float* p0 = xb + koff * HW + (tv0 ? col0 : m16);
    const float* p1 = xb + koff * HW + (tv1 ? col1 : m16);
    const float* p2 = xb + koff * HW + (tv2 ? col2 : m16);
    const float* p3 = xb + koff * HW + (tv3 ? col3 : m16);

    v8f acc0 = {}, acc1 = {}, acc2 = {}, acc3 = {};

    #pragma unroll 4
    for (int k = 0; k < 128; k += 4) {
        // A fragment: A(m, k..k+3); lane m16 holds (k+koff, k+koff+1)
        v2f a;
        a.x = ldsW[m16][k + koff];
        a.y = ldsW[m16][k + koff + 1];

        // B fragments: rows k+koff and k+koff+1 at this lane's column
        v2f b0, b1, b2, b3;
        b0.x = p0[0]; b0.y = p0[HW];
        b1.x = p1[0]; b1.y = p1[HW];
        b2.x = p2[0]; b2.y = p2[HW];
        b3.x = p3[0]; b3.y = p3[HW];
        p0 += 4 * HW; p1 += 4 * HW; p2 += 4 * HW; p3 += 4 * HW;

        acc0 = __builtin_amdgcn_wmma_f32_16x16x4_f32(false, a, false, b0, (short)0, acc0, false, false);
        acc1 = __builtin_amdgcn_wmma_f32_16x16x4_f32(false, a, false, b1, (short)0, acc1, false, false);
        acc2 = __builtin_amdgcn_wmma_f32_16x16x4_f32(false, a, false, b2, (short)0, acc2, false, false);
        acc3 = __builtin_amdgcn_wmma_f32_16x16x4_f32(false, a, false, b3, (short)0, acc3, false, false);
    }

    // ---- store results: C/D layout: VGPR r = row (r + hi*8), lane m16 = column ----
    const size_t ybase = (size_t)(b * 128 + o0 + hi * 8) * HW;
    if (tv0) {
        float* yp = y + ybase + col0;
        #pragma unroll
        for (int r = 0; r < 8; ++r) yp[r * HW] = acc0[r];
    }
    if (tv1) {
        float* yp = y + ybase + col1;
        #pragma unroll
        for (int r = 0; r < 8; ++r) yp[r * HW] = acc1[r];
    }
    if (tv2) {
        float* yp = y + ybase + col2;
        #pragma unroll
        for (int r = 0; r < 8; ++r) yp[r * HW] = acc2[r];
    }
    if (tv3) {
        float* yp = y + ybase + col3;
        #pragma unroll
        for (int r = 0; r < 8; ++r) yp[r * HW] = acc3[r];
    }

    // ---- instance-norm statistics: per-row sum and sum of squares ----
    if constexpr (DOSTATS) {
        const float f0 = tv0 ? 1.0f : 0.0f;
        const float f1 = tv1 ? 1.0f : 0.0f;
        const float f2 = tv2 ? 1.0f : 0.0f;
        const float f3 = tv3 ? 1.0f : 0.0f;
        #pragma unroll
        for (int r = 0; r < 8; ++r) {
            float sv, qv;
            if constexpr (FULL) {
                sv = acc0[r] + acc1[r] + acc2[r] + acc3[r];
                qv = acc0[r] * acc0[r] + acc1[r] * acc1[r]
                   + acc2[r] * acc2[r] + acc3[r] * acc3[r];
            } else {
                sv = acc0[r] * f0 + acc1[r] * f1 + acc2[r] * f2 + acc3[r] * f3;
                qv = acc0[r] * acc0[r] * f0 + acc1[r] * acc1[r] * f1
                   + acc2[r] * acc2[r] * f2 + acc3[r] * acc3[r] * f3;
            }
            // reduce across the 16 lanes of each half (masks <=8 stay in-half)
            #pragma unroll
            for (int off = 8; off >= 1; off >>= 1) {
                sv += __shfl_xor(sv, off, 32);
                qv += __shfl_xor(qv, off, 32);
            }
            if (m16 == 0) {
                int bo = b * 128 + o0 + r + hi * 8;
                atomicAdd(&stats[bo * 2 + 0], sv);
                atomicAdd(&stats[bo * 2 + 1], qv);
            }
        }
    }
}

// ---------------------------------------------------------------------------
// (sum, sumsq) -> (mean, rsqrt(var + eps)) per (b, o); biased variance.
// ---------------------------------------------------------------------------
__global__ void finalize_stats(float* __restrict__ stats, float invHW) {
    int i = blockIdx.x * blockDim.x + threadIdx.x;
    if (i < 2048) {
        float s = stats[i * 2 + 0];
        float q = stats[i * 2 + 1];
        float mean = s * invHW;
        float var  = q * invHW - mean * mean;
        stats[i * 2 + 0] = mean;
        stats[i * 2 + 1] = rsqrtf(var + EPS);
    }
}

// ---------------------------------------------------------------------------
// y = (y - mean) * rinv, streamed as float4.  HW >= 64 so a float4 never
// straddles a (b,o) row; SHIFT4 = log2(HW) - 2 maps float4 index -> (b,o).
// ---------------------------------------------------------------------------
template<int SHIFT4>
__global__ void in_norm(float4* __restrict__ y, const float* __restrict__ stats,
                        long long n4) {
    long long i = (long long)blockIdx.x * blockDim.x + threadIdx.x;
    long long stride = (long long)gridDim.x * blockDim.x;
    for (; i < n4; i += stride) {
        int bo = (int)(i >> SHIFT4);
        float m = stats[bo * 2 + 0];
        float r = stats[bo * 2 + 1];
        float4 v = y[i];
        v.x = (v.x - m) * r;
        v.y = (v.y - m) * r;
        v.z = (v.z - m) * r;
        v.w = (v.w - m) * r;
        y[i] = v;
    }
}

// ---------------------------------------------------------------------------
extern "C" void kernel_launch(void* const* d_in, const int* in_sizes, int n_in,
                              void* d_out, int out_size, void* d_ws, size_t ws_size,
                              hipStream_t stream) {
    (void)in_sizes; (void)n_in; (void)out_size; (void)ws_size;

    // setup_inputs order: x1, l1fs, x2, l2fs, x3, l3fs, x4, l4fs, x5, l5fs
    const float* xs[5] = { (const float*)d_in[0], (const float*)d_in[2],
                           (const float*)d_in[4], (const float*)d_in[6],
                           (const float*)d_in[8] };
    const float* ws_[5] = { (const float*)d_in[1], (const float*)d_in[3],
                            (const float*)d_in[5], (const float*)d_in[7],
                            (const float*)d_in[9] };

    const int HWs[5] = { 16384, 4096, 1024, 256, 64 };

    float* out   = (float*)d_out;
    float* stats = (float*)d_ws;   // 4 layers * 2048 * 2 floats = 64 KB

    // output offsets (outputs concatenated flat in return order)
    size_t offs[5];
    size_t acc = 0;
    for (int L = 0; L < 5; ++L) {
        offs[L] = acc;
        acc += (size_t)16 * 128 * (size_t)HWs[L];
    }

    // 1) zero the stats workspace
    zero_stats<<<(16384 + 255) / 256, 256, 0, stream>>>(stats, 16384);

    // 2) batched WMMA GEMMs (with fused stat accumulation for layers 1..4)
    dynconv_gemm<16384, true, true><<<dim3(64, 8, 16), 128, 0, stream>>>(
        xs[0], ws_[0], out + offs[0], stats + 0 * 4096);
    dynconv_gemm<4096, true, true><<<dim3(16, 8, 16), 128, 0, stream>>>(
        xs[1], ws_[1], out + offs[1], stats + 1 * 4096);
    dynconv_gemm<1024, true, true><<<dim3(4, 8, 16), 128, 0, stream>>>(
        xs[2], ws_[2], out + offs[2], stats + 2 * 4096);
    dynconv_gemm<256, true, true><<<dim3(1, 8, 16), 128, 0, stream>>>(
        xs[3], ws_[3], out + offs[3], stats + 3 * 4096);
    dynconv_gemm<64, false, false><<<dim3(1, 8, 16), 128, 0, stream>>>(
        xs[4], ws_[4], out + offs[4], stats);          // stats unused

    // 3) finalize stats + normalize layers 1..4 (layer 5 is not normalized)
    {
        long long n4;
        finalize_stats<<<8, 256, 0, stream>>>(stats + 0 * 4096, 1.0f / 16384.0f);
        n4 = ((long long)16 * 128 * 16384) >> 2;
        in_norm<12><<<(int)((n4 + 255) / 256), 256, 0, stream>>>(
            (float4*)(out + offs[0]), stats + 0 * 4096, n4);

        finalize_stats<<<8, 256, 0, stream>>>(stats + 1 * 4096, 1.0f / 4096.0f);
        n4 = ((long long)16 * 128 * 4096) >> 2;
        in_norm<10><<<(int)((n4 + 255) / 256), 256, 0, stream>>>(
            (float4*)(out + offs[1]), stats + 1 * 4096, n4);

        finalize_stats<<<8, 256, 0, stream>>>(stats + 2 * 4096, 1.0f / 1024.0f);
        n4 = ((long long)16 * 128 * 1024) >> 2;
        in_norm<8><<<(int)((n4 + 255) / 256), 256, 0, stream>>>(
            (float4*)(out + offs[2]), stats + 2 * 4096, n4);

        finalize_stats<<<8, 256, 0, stream>>>(stats + 3 * 4096, 1.0f / 256.0f);
        n4 = ((long long)16 * 128 * 256) >> 2;
        in_norm<6><<<(int)((n4 + 255) / 256), 256, 0, stream>>>(
            (float4*)(out + offs[3]), stats + 3 * 4096, n4);
    }
}